// TreeDecoderTeacherForced_16458314678317
// MI455X (gfx1250) — compile-verified
//
#include <hip/hip_runtime.h>
#include <hip/hip_bf16.h>

// ---------------------------------------------------------------------------
// TreeDecoderTeacherForced for MI455X (gfx1250, wave32, WMMA f32 16x16x4)
//
// Teacher-forced routing degenerates (keys == arange): the child-MLP output
// reshaped (4N,64) IS the next level's h. Morton neighbor indices computed
// in-register. Weights repacked once into float2 (k,k+1)-pairs so each B
// fragment is one coalesced b64 load.
//
// RT=2 row tiling: 32 rows per wave = 8 v8f accumulators (64 VGPRs), which
// amortizes each B load over 8 WMMAs while leaving the compiler enough VGPR
// headroom to pipeline the K-loop WITHOUT scratch spills (RT=4 spilled).
// ---------------------------------------------------------------------------

typedef float v2f __attribute__((ext_vector_type(2)));
typedef float v8f __attribute__((ext_vector_type(8)));

#define NVAL_TOTAL 87381      // sum 4^d, d=0..8
#define LVL8_N     65536

static __device__ __forceinline__ v8f wmma_f32_k4(v2f a, v2f b, v8f c) {
  return __builtin_amdgcn_wmma_f32_16x16x4_f32(false, a, false, b, (short)0, c,
                                               false, false);
}

static __device__ __forceinline__ v8f vzero8() {
  v8f v = {0.f, 0.f, 0.f, 0.f, 0.f, 0.f, 0.f, 0.f};
  return v;
}

static __device__ __forceinline__ unsigned mdeint(unsigned v) {
  v &= 0x55555555u;
  v = (v | (v >> 1)) & 0x33333333u;
  v = (v | (v >> 2)) & 0x0F0F0F0Fu;
  v = (v | (v >> 4)) & 0x00FF00FFu;
  v = (v | (v >> 8)) & 0x0000FFFFu;
  return v;
}
static __device__ __forceinline__ unsigned minter(unsigned v) {
  v &= 0xFFFFu;
  v = (v | (v << 8)) & 0x00FF00FFu;
  v = (v | (v << 4)) & 0x0F0F0F0Fu;
  v = (v | (v << 2)) & 0x33333333u;
  v = (v | (v << 1)) & 0x55555555u;
  return v;
}

// (32 x K) LDS tile  @  K x [ncol0..ncol0+64) of packed weights.
// Wp layout: Wp[pr*ldw + n] = { W[2*pr][n], W[2*pr+1][n] }  (pr = packed row)
// Per K-chunk: 4 coalesced b64 B loads shared by 2 row sub-tiles (8 wmma).
static __device__ __forceinline__ void gemm_rt2(const float* __restrict__ A, int lda,
                                                const v2f* __restrict__ Wp, int ldw,
                                                int ncol0, int K, v8f acc[2][4]) {
  const int lane = threadIdx.x & 31;
  const int half = lane >> 4;   // 0: rows k..k+1, 1: rows k+2..k+3
  const int l15  = lane & 15;
  for (int k = 0; k < K; k += 4) {
    const int pr = (k >> 1) + half;         // packed row index
    v2f b[4];
#pragma unroll
    for (int nb = 0; nb < 4; ++nb)
      b[nb] = Wp[(size_t)pr * ldw + ncol0 + nb * 16 + l15];
    const int kk = k + 2 * half;
#pragma unroll
    for (int rt = 0; rt < 2; ++rt) {
      const v2f a = *(const v2f*)(A + (rt * 16 + l15) * lda + kk);
#pragma unroll
      for (int nb = 0; nb < 4; ++nb)
        acc[rt][nb] = wmma_f32_k4(a, b[nb], acc[rt][nb]);
    }
  }
}

static __device__ __forceinline__ void acc_zero(v8f acc[2][4]) {
#pragma unroll
  for (int i = 0; i < 2; ++i)
#pragma unroll
    for (int j = 0; j < 4; ++j) acc[i][j] = vzero8();
}

static __device__ __forceinline__ void acc_store_lds(const v8f acc[2][4], float* out,
                                                     int lds, const float* __restrict__ bias,
                                                     bool relu) {
  const int lane = threadIdx.x & 31;
  const int half = lane >> 4, l15 = lane & 15;
#pragma unroll
  for (int rt = 0; rt < 2; ++rt)
#pragma unroll
    for (int nb = 0; nb < 4; ++nb)
#pragma unroll
      for (int r = 0; r < 8; ++r) {
        const int row = rt * 16 + r + 8 * half;
        const int col = nb * 16 + l15;
        float v = acc[rt][nb][r] + bias[col];
        if (relu) v = fmaxf(v, 0.f);
        out[row * lds + col] = v;
      }
}

static __device__ __forceinline__ void acc_store_global(const v8f acc[2][4],
                                                        float* __restrict__ out, int ostride,
                                                        int col0, int m0, int N,
                                                        const float* __restrict__ bias) {
  const int lane = threadIdx.x & 31;
  const int half = lane >> 4, l15 = lane & 15;
#pragma unroll
  for (int rt = 0; rt < 2; ++rt)
#pragma unroll
    for (int nb = 0; nb < 4; ++nb)
#pragma unroll
      for (int r = 0; r < 8; ++r) {
        const int row = rt * 16 + r + 8 * half;
        const int node = m0 + row;
        const int col = col0 + nb * 16 + l15;
        if (node < N) out[(size_t)node * ostride + col] = acc[rt][nb][r] + bias[col];
      }
}

// ------------------------------- K1: fuse ----------------------------------
// X: 32 rows x 172 floats (172 mod 64 = 44, gcd 4 -> 16 distinct banks;
//    multiple of 4 -> float4 staging). Y aliases X after GEMM1.
__global__ void __launch_bounds__(32)
k_fuse(const float* __restrict__ h_src, int h_stride,
       const float* __restrict__ emb,
       const float* __restrict__ ln_g, const float* __restrict__ ln_b,
       const v2f* __restrict__ W1p /*packed 168x64*/,
       const float* __restrict__ b1,
       const v2f* __restrict__ W2p /*packed 64x64*/,
       const float* __restrict__ b2,
       float* __restrict__ h1_out, int N, int depth) {
  __shared__ float X[32 * 172];
  const int m0 = blockIdx.x * 32;
  const int lane = threadIdx.x;

  // ---- stage h part (cols 0..63), float4-coalesced ----
  for (int idx = lane; idx < 32 * 16; idx += 32) {
    const int row = idx >> 4, c4 = idx & 15;
    const int node = m0 + row;
    float4 v = {0.f, 0.f, 0.f, 0.f};
    if (node < N) v = ((const float4*)(h_src + (size_t)node * h_stride))[c4];
    *(float4*)&X[row * 172 + c4 * 4] = v;
  }
  // ---- LayerNorm(emb) -> cols 64..127 (one row per lane) ----
  {
    const int row = lane;
    const int node = m0 + row;
    if (node < N) {
      const float* e = emb + (size_t)node * 64;
      float mu = 0.f;
      for (int c = 0; c < 64; ++c) mu += e[c];
      mu *= (1.f / 64.f);
      float var = 0.f;
      for (int c = 0; c < 64; ++c) { float t = e[c] - mu; var += t * t; }
      var *= (1.f / 64.f);
      const float inv = rsqrtf(var + 1e-5f);
      for (int c = 0; c < 64; ++c)
        X[row * 172 + 64 + c] = (e[c] - mu) * inv * ln_g[c] + ln_b[c];
    } else {
      for (int c = 0; c < 64; ++c) X[row * 172 + 64 + c] = 0.f;
    }
  }
  // ---- fourier pos -> cols 128..166, col 167 zero pad ----
  {
    const int row = lane;
    const int node = m0 + row;
    float* P = &X[row * 172 + 128];
    if (node < N) {
      const float res = (float)(1 << depth);
      float p[3];
      p[0] = ((float)mdeint((unsigned)node) + 0.5f) / res;
      p[1] = ((float)mdeint(((unsigned)node) >> 1) + 0.5f) / res;
      p[2] = (float)depth * 0.125f;
      P[0] = p[0]; P[1] = p[1]; P[2] = p[2];
      const float two_pi = 6.28318530717958647692f;
#pragma unroll
      for (int dd = 0; dd < 3; ++dd) {
        const float base = p[dd] * two_pi;
#pragma unroll
        for (int f = 0; f < 6; ++f) {
          const float ang = base * (float)(1 << f);
          P[3 + dd * 12 + f]     = __sinf(ang);
          P[3 + dd * 12 + 6 + f] = __cosf(ang);
        }
      }
      P[39] = 0.f;
    } else {
      for (int c = 0; c < 40; ++c) P[c] = 0.f;
    }
  }
  __syncthreads();

  // ---- GEMM1: (32x168) @ (168x64), relu ----
  v8f acc[2][4];
  acc_zero(acc);
  gemm_rt2(X, 172, W1p, 64, 0, 168, acc);
  __syncthreads();               // X fully consumed; alias Y onto X
  float* Y = X;                  // 32 x 68
  acc_store_lds(acc, Y, 68, b1, true);
  __syncthreads();

  // ---- GEMM2: (32x64) @ (64x64) -> h1 ----
  acc_zero(acc);
  gemm_rt2(Y, 68, W2p, 64, 0, 64, acc);
  acc_store_global(acc, h1_out, 64, 0, m0, N, b2);
}

// --------------------------- K2: conv + heads ------------------------------
__global__ void __launch_bounds__(32)
k_conv_heads(const float* __restrict__ h1,
             const v2f* __restrict__ convWp /*packed 576x64 slab*/,
             const float* __restrict__ convb,
             const v2f* __restrict__ vW1p, const float* __restrict__ vb1,
             const float* __restrict__ vW2, const float* __restrict__ vb2,
             const v2f* __restrict__ sW1p, const float* __restrict__ sb1,
             const float* __restrict__ sW2, const float* __restrict__ sb2,
             const v2f* __restrict__ cW1p, const float* __restrict__ cb1,
             const v2f* __restrict__ cW2p /*packed 64x256*/,
             const float* __restrict__ cb2,
             float* __restrict__ val_out,
             float* __restrict__ split_out,
             float* __restrict__ h_next,
             int N, int depth, int has_children) {
  __shared__ float Aj[32 * 68];
  __shared__ float H2[32 * 68];
  __shared__ float Y[32 * 68];
  const int m0 = blockIdx.x * 32;
  const int lane = threadIdx.x;
  const int res = 1 << depth;

  v8f acc[2][4];
  acc_zero(acc);

  // ---- conv: sum_j gather_j(h1) @ convW[j*64 .. j*64+63] ----
  for (int j = 0; j < 9; ++j) {
    const int dx = j % 3 - 1, dy = j / 3 - 1;
    {
      const int row = lane;            // one row per lane
      const int node = m0 + row;
      int nidx = -1;
      if (node < N) {
        const int x = (int)mdeint((unsigned)node) + dx;
        const int y = (int)mdeint(((unsigned)node) >> 1) + dy;
        if (x >= 0 && x < res && y >= 0 && y < res)
          nidx = (int)(minter((unsigned)x) | (minter((unsigned)y) << 1));
      }
      float4* d4 = (float4*)&Aj[row * 68];
      if (nidx >= 0) {
        const float4* s4 = (const float4*)&h1[(size_t)nidx * 64];
#pragma unroll
        for (int c = 0; c < 16; ++c) d4[c] = s4[c];
      } else {
        const float4 z = {0.f, 0.f, 0.f, 0.f};
#pragma unroll
        for (int c = 0; c < 16; ++c) d4[c] = z;
      }
    }
    __syncthreads();
    gemm_rt2(Aj, 68, convWp + (size_t)j * 32 * 64, 64, 0, 64, acc);
    __syncthreads();   // before restaging Aj
  }
  acc_store_lds(acc, H2, 68, convb, false);
  __syncthreads();

  // ---- val head ----
  acc_zero(acc);
  gemm_rt2(H2, 68, vW1p, 64, 0, 64, acc);
  acc_store_lds(acc, Y, 68, vb1, true);
  __syncthreads();
  {
    const int row = lane;
    const int node = m0 + row;
    if (node < N) {
      float s = vb2[0];
      for (int c = 0; c < 64; ++c) s += Y[row * 68 + c] * vW2[c];
      val_out[node] = s;
    }
  }
  __syncthreads();

  if (has_children) {
    // ---- split head ----
    acc_zero(acc);
    gemm_rt2(H2, 68, sW1p, 64, 0, 64, acc);
    acc_store_lds(acc, Y, 68, sb1, true);
    __syncthreads();
    {
      const int row = lane;
      const int node = m0 + row;
      if (node < N) {
        float s = sb2[0];
        for (int c = 0; c < 64; ++c) s += Y[row * 68 + c] * sW2[c];
        split_out[node] = s;
      }
    }
    __syncthreads();

    // ---- child head: Y = relu(H2@cW1+cb1); h' = Y@cW2+cb2 (32x256) ----
    acc_zero(acc);
    gemm_rt2(H2, 68, cW1p, 64, 0, 64, acc);
    acc_store_lds(acc, Y, 68, cb1, true);
    __syncthreads();
    for (int g = 0; g < 4; ++g) {
      v8f a2[2][4];
      acc_zero(a2);
      gemm_rt2(Y, 68, cW2p, 256, g * 64, 64, a2);
      // cf[node][q][c] == next-level h[node*4+q][c] (contiguous 256/row)
      acc_store_global(a2, h_next, 256, g * 64, m0, N, cb2);
    }
  }
}

// Repack W (Ksrc x ncols, zero-padded to Kpad rows) into float2 k-pairs:
//   Wp[pr*ncols + n] = { W[2pr][n], W[2pr+1][n] }
__global__ void k_repack(const float* __restrict__ W, v2f* __restrict__ Wp,
                         int Ksrc, int Kpad, int ncols) {
  const int e = blockIdx.x * 256 + threadIdx.x;
  const int total = (Kpad >> 1) * ncols;
  if (e >= total) return;
  const int pr = e / ncols;
  const int n = e - pr * ncols;
  const int r0 = 2 * pr, r1 = 2 * pr + 1;
  v2f v;
  v.x = (r0 < Ksrc) ? W[(size_t)r0 * ncols + n] : 0.f;
  v.y = (r1 < Ksrc) ? W[(size_t)r1 * ncols + n] : 0.f;
  Wp[e] = v;
}

// ---------------------------------------------------------------------------
// Input order (setup_inputs dict, tuples flattened to leaves):
//   [0..8] keys (unused) [9..17] neighs (unused) [18..26] parent (unused)
//   [27..35] quad (unused) [36..44] emb_list [45] root [46] ln_g [47] ln_b
//   [48..51] fuse_W1,b1,W2,b2 [52..55] split_* [56..59] child_*
//   [60..63] val_* [64] conv_W (9,576,64) [65] conv_b (9,64)
// d_out: val (87381) ++ split (21845) floats.
// ---------------------------------------------------------------------------
extern "C" void kernel_launch(void* const* d_in, const int* in_sizes, int n_in,
                              void* d_out, int out_size, void* d_ws, size_t ws_size,
                              hipStream_t stream) {
  (void)in_sizes; (void)n_in; (void)out_size; (void)ws_size;

  const float* root    = (const float*)d_in[45];
  const float* ln_g    = (const float*)d_in[46];
  const float* ln_b    = (const float*)d_in[47];
  const float* fuse_W1 = (const float*)d_in[48];
  const float* fuse_b1 = (const float*)d_in[49];
  const float* fuse_W2 = (const float*)d_in[50];
  const float* fuse_b2 = (const float*)d_in[51];
  const float* sW1 = (const float*)d_in[52];
  const float* sb1 = (const float*)d_in[53];
  const float* sW2 = (const float*)d_in[54];
  const float* sb2 = (const float*)d_in[55];
  const float* cW1 = (const float*)d_in[56];
  const float* cb1 = (const float*)d_in[57];
  const float* cW2 = (const float*)d_in[58];
  const float* cb2 = (const float*)d_in[59];
  const float* vW1 = (const float*)d_in[60];
  const float* vb1 = (const float*)d_in[61];
  const float* vW2 = (const float*)d_in[62];
  const float* vb2 = (const float*)d_in[63];
  const float* conv_W = (const float*)d_in[64];
  const float* conv_b = (const float*)d_in[65];

  // -------- workspace layout (floats; all offsets even for v2f) --------
  float* ws = (float*)d_ws;
  float* hbuf   = ws;                                  // 65536*64
  float* h1buf  = hbuf  + (size_t)LVL8_N * 64;         // 65536*64
  float* fw1p   = h1buf + (size_t)LVL8_N * 64;         // (168/2)*64*2 = 10752
  float* fw2p   = fw1p  + 10752;                       // (64/2)*64*2  =  4096
  float* convp  = fw2p  + 4096;                        // (5184/2)*64*2 = 331776
  float* vw1p   = convp + 331776;                      //  4096
  float* sw1p   = vw1p  + 4096;                        //  4096
  float* cw1p   = sw1p  + 4096;                        //  4096
  float* cw2p   = cw1p  + 4096;                        // (64/2)*256*2 = 16384

  // -------- one-time weight repacks --------
  auto rp = [&](const float* W, float* Wp, int Ksrc, int Kpad, int ncols) {
    const int total = (Kpad >> 1) * ncols;
    k_repack<<<dim3((total + 255) / 256), dim3(256), 0, stream>>>(
        W, (v2f*)Wp, Ksrc, Kpad, ncols);
  };
  rp(fuse_W1, fw1p, 167, 168, 64);
  rp(fuse_W2, fw2p, 64, 64, 64);
  rp(conv_W,  convp, 9 * 576, 9 * 576, 64);   // slabs are even-height: pairs never cross
  rp(vW1, vw1p, 64, 64, 64);
  rp(sW1, sw1p, 64, 64, 64);
  rp(cW1, cw1p, 64, 64, 64);
  rp(cW2, cw2p, 64, 64, 256);

  float* val_base   = (float*)d_out;
  float* split_base = val_base + NVAL_TOTAL;

  for (int d = 0; d <= 8; ++d) {
    const int N = 1 << (2 * d);
    const int tiles = (N + 31) / 32;
    const int voff = (N - 1) / 3;  // sum of 4^k, k<d

    k_fuse<<<dim3(tiles), dim3(32), 0, stream>>>(
        d == 0 ? root : hbuf, d == 0 ? 0 : 64,
        (const float*)d_in[36 + d], ln_g, ln_b,
        (const v2f*)fw1p, fuse_b1, (const v2f*)fw2p, fuse_b2,
        h1buf, N, d);

    k_conv_heads<<<dim3(tiles), dim3(32), 0, stream>>>(
        h1buf,
        (const v2f*)convp + (size_t)d * 288 * 64, conv_b + d * 64,
        (const v2f*)vw1p, vb1, vW2, vb2,
        (const v2f*)sw1p, sb1, sW2, sb2,
        (const v2f*)cw1p, cb1, (const v2f*)cw2p, cb2,
        val_base + voff, split_base + voff,
        hbuf, N, d, (d < 8) ? 1 : 0);
  }
}